// MergeDecoder_25168508354597
// MI455X (gfx1250) — compile-verified
//
#include <hip/hip_runtime.h>

#define F_    2048
#define C_    64
#define NELEM (C_ * F_)
#define SEG   8          // split-K factor for the child GEMV
#define FSEG  (F_ / SEG) // 256 f-rows per segment

typedef __attribute__((ext_vector_type(2))) float v2f;
typedef __attribute__((ext_vector_type(8))) float v8f;

// ---------------------------------------------------------------------------
// Kernel 1a: split-f partial child GEMV.
// partial[seg][c][o] = sum_{f in seg} p[f] * W[c,f,o]
// grid = C * 2 * SEG = 1024 blocks x 256 threads (8192 waves) so enough
// b128 loads are in flight to saturate 23.3 TB/s.  global_prefetch_b8 runs
// 128 rows ahead (doesn't consume LOADcnt).
// ---------------------------------------------------------------------------
__global__ void __launch_bounds__(256)
child_mlp_partial_kernel(const float* __restrict__ p,
                         const float* __restrict__ W,
                         float* __restrict__ partial) {
    __shared__ float pl[FSEG];
    const int t = threadIdx.x;
    int b = blockIdx.x;
    const int seg = b & (SEG - 1);
    b >>= 3;
    const int c = b >> 1;
    const int o = ((b & 1) << 10) + (t << 2);
    const int fbase = seg << 8;

    pl[t] = p[fbase + t];          // 256 threads load the 256-row segment
    __syncthreads();

    const size_t wbase = ((size_t)c * F_) * F_ + o;
    float4 acc = {0.f, 0.f, 0.f, 0.f};

    for (int f0 = 0; f0 < FSEG; f0 += 16) {
        int pfr = fbase + f0 + 128;            // prefetch distance: 128 rows
        if (pfr > F_ - 1) pfr = F_ - 1;        // stay inside this child's W
        __builtin_prefetch(W + wbase + (size_t)pfr * F_, 0, 1);
        #pragma unroll
        for (int u = 0; u < 16; ++u) {
            const int f = f0 + u;
            const float pv = pl[f];
            const float4 w = *(const float4*)(W + wbase + (size_t)(fbase + f) * F_);
            acc.x = fmaf(pv, w.x, acc.x);
            acc.y = fmaf(pv, w.y, acc.y);
            acc.z = fmaf(pv, w.z, acc.z);
            acc.w = fmaf(pv, w.w, acc.w);
        }
    }
    *(float4*)(partial + (size_t)seg * NELEM + (size_t)c * F_ + o) = acc;
}

// ---------------------------------------------------------------------------
// Kernel 1b: reduce the SEG partials, add bias, relu.
// out[c,o] = relu(b[c,o] + sum_s partial[s][c][o]); fixed order -> deterministic.
// ---------------------------------------------------------------------------
__global__ void __launch_bounds__(256)
child_reduce_kernel(const float* __restrict__ partial,
                    const float* __restrict__ bias,
                    float* __restrict__ out) {
    const int i4 = blockIdx.x * 256 + threadIdx.x;     // float4 index
    float4 acc = ((const float4*)bias)[i4];
    #pragma unroll
    for (int s = 0; s < SEG; ++s) {
        const float4 v = ((const float4*)partial)[(size_t)s * (NELEM / 4) + i4];
        acc.x += v.x; acc.y += v.y; acc.z += v.z; acc.w += v.w;
    }
    float4 r;
    r.x = fmaxf(acc.x, 0.f); r.y = fmaxf(acc.y, 0.f);
    r.z = fmaxf(acc.z, 0.f); r.w = fmaxf(acc.w, 0.f);
    ((float4*)out)[i4] = r;
}

// ---------------------------------------------------------------------------
// Kernel 1 (fallback, if ws too small for partials): single-pass child GEMV.
// ---------------------------------------------------------------------------
__global__ void __launch_bounds__(256)
child_mlp_kernel(const float* __restrict__ p,
                 const float* __restrict__ W,
                 const float* __restrict__ b,
                 float* __restrict__ out) {
    __shared__ __align__(16) float pl[F_];
    const int t = threadIdx.x;
    const float4* p4 = (const float4*)p;
    ((float4*)pl)[t]       = p4[t];
    ((float4*)pl)[t + 256] = p4[t + 256];
    __syncthreads();

    const int c = blockIdx.x >> 1;
    const int o = ((blockIdx.x & 1) << 10) + (t << 2);
    const size_t wbase = ((size_t)c * F_) * F_ + o;

    float4 acc = *(const float4*)(b + (size_t)c * F_ + o);
    for (int f0 = 0; f0 < F_; f0 += 16) {
        int pfr = f0 + 128;
        if (pfr > F_ - 1) pfr = F_ - 1;
        __builtin_prefetch(W + wbase + (size_t)pfr * F_, 0, 1);
        #pragma unroll
        for (int u = 0; u < 16; ++u) {
            const float pv = pl[f0 + u];
            const float4 w = *(const float4*)(W + wbase + (size_t)(f0 + u) * F_);
            acc.x = fmaf(pv, w.x, acc.x);
            acc.y = fmaf(pv, w.y, acc.y);
            acc.z = fmaf(pv, w.z, acc.z);
            acc.w = fmaf(pv, w.w, acc.w);
        }
    }
    float4 r;
    r.x = fmaxf(acc.x, 0.f); r.y = fmaxf(acc.y, 0.f);
    r.z = fmaxf(acc.z, 0.f); r.w = fmaxf(acc.w, 0.f);
    *(float4*)(out + (size_t)c * F_ + o) = r;
}

// ---------------------------------------------------------------------------
// Kernel 2: GIN aggregation. h[c,:] = x[c,:] + (c>0 ? colsum(x) : 0)
// ---------------------------------------------------------------------------
__global__ void __launch_bounds__(256)
gin_agg_kernel(const float* __restrict__ x, float* __restrict__ h) {
    const int col = blockIdx.x * 256 + threadIdx.x;
    float s = 0.f;
    #pragma unroll
    for (int c = 0; c < C_; ++c) s += x[c * F_ + col];
    h[col] = x[col];
    #pragma unroll
    for (int c = 1; c < C_; ++c) h[c * F_ + col] = x[c * F_ + col] + s;
}

// ---------------------------------------------------------------------------
// Kernel 3: D[64,2048] = act(A[64,2048] @ W[2048,2048] + bias) via
// V_WMMA_F32_16X16X4_F32 (codegen-confirmed last round: clean pipelined
// wmma + ds_load_2addr_b64 + clause-grouped global B loads).
// ---------------------------------------------------------------------------
#define GK   2048
#define GN   2048
#define CK   128
#define LDSK 132

__global__ void __launch_bounds__(128)
gemm_wmma_kernel(const float* __restrict__ A,
                 const float* __restrict__ W,
                 const float* __restrict__ bias,
                 float* __restrict__ D,
                 int do_relu) {
    __shared__ __align__(16) float As[64 * LDSK];

    const int tid   = threadIdx.x;
    const int wave  = tid >> 5;
    const int lane  = tid & 31;
    const int m0    = wave << 4;
    const int n0    = blockIdx.x << 4;
    const int row16 = lane & 15;
    const int khalf = (lane >> 4) << 1;

    v8f acc = {};

    for (int k0 = 0; k0 < GK; k0 += CK) {
        #pragma unroll
        for (int i = 0; i < 16; ++i) {
            const int idx = tid + (i << 7);
            const int r   = idx >> 5;
            const int c4  = idx & 31;
            const float4 v = *(const float4*)(A + (size_t)r * GK + k0 + (c4 << 2));
            *(float4*)&As[r * LDSK + (c4 << 2)] = v;
        }
        __syncthreads();

        const float* wp = W + (size_t)(k0 + khalf) * GN + n0 + row16;
        const float* ap = &As[(m0 + row16) * LDSK + khalf];

        #pragma unroll
        for (int kk = 0; kk < CK; kk += 4) {
            v2f a, bf;
            a.x  = ap[kk];
            a.y  = ap[kk + 1];
            bf.x = wp[(size_t)kk * GN];
            bf.y = wp[(size_t)(kk + 1) * GN];
            acc = __builtin_amdgcn_wmma_f32_16x16x4_f32(
                      false, a, false, bf, (short)0, acc, false, false);
        }
        __syncthreads();
    }

    const float bv  = bias[n0 + row16];
    const int   mhi = (lane >> 4) << 3;
    #pragma unroll
    for (int r = 0; r < 8; ++r) {
        float v = acc[r] + bv;
        if (do_relu) v = fmaxf(v, 0.f);
        D[(size_t)(m0 + r + mhi) * GN + n0 + row16] = v;
    }
}

// ---------------------------------------------------------------------------
// Kernel 4: BatchNorm1d (training mode, biased var) over the node axis.
// ---------------------------------------------------------------------------
__global__ void __launch_bounds__(256)
bn_kernel(const float* __restrict__ x,
          const float* __restrict__ g,
          const float* __restrict__ beta,
          float* __restrict__ y) {
    const int col = blockIdx.x * 256 + threadIdx.x;
    float s = 0.f, ss = 0.f;
    #pragma unroll
    for (int c = 0; c < C_; ++c) {
        const float v = x[c * F_ + col];
        s += v;
        ss = fmaf(v, v, ss);
    }
    const float m    = s * (1.f / C_);
    const float var  = ss * (1.f / C_) - m * m;
    const float inv  = rsqrtf(var + 1e-5f);
    const float gg   = g[col] * inv;
    const float bb   = beta[col] - gg * m;
    #pragma unroll
    for (int c = 0; c < C_; ++c)
        y[c * F_ + col] = fmaf(gg, x[c * F_ + col], bb);
}

// ---------------------------------------------------------------------------
extern "C" void kernel_launch(void* const* d_in, const int* in_sizes, int n_in,
                              void* d_out, int out_size, void* d_ws, size_t ws_size,
                              hipStream_t stream) {
    const float* parent = (const float*)d_in[0];
    const float* childW = (const float*)d_in[1];
    const float* childb = (const float*)d_in[2];
    const float* W1a    = (const float*)d_in[3];
    const float* b1a    = (const float*)d_in[4];
    const float* W1b    = (const float*)d_in[5];
    const float* b1b    = (const float*)d_in[6];
    const float* g1     = (const float*)d_in[7];
    const float* be1    = (const float*)d_in[8];
    const float* W2a    = (const float*)d_in[9];
    const float* b2a    = (const float*)d_in[10];
    const float* W2b    = (const float*)d_in[11];
    const float* b2b    = (const float*)d_in[12];
    const float* g2     = (const float*)d_in[13];
    const float* be2    = (const float*)d_in[14];
    float* out  = (float*)d_out;

    float* buf0 = (float*)d_ws;
    float* buf1 = buf0 + NELEM;
    float* buf2 = buf1 + NELEM;
    float* pbuf = buf2 + NELEM;   // SEG * NELEM partials

    const size_t need_split = (size_t)(3 + SEG) * NELEM * sizeof(float);

    // x0 = relu(parent @ child_W + child_b)            [64,2048]
    if (ws_size >= need_split) {
        child_mlp_partial_kernel<<<C_ * 2 * SEG, 256, 0, stream>>>(parent, childW, pbuf);
        child_reduce_kernel<<<NELEM / 4 / 256, 256, 0, stream>>>(pbuf, childb, buf0);
    } else {
        child_mlp_kernel<<<C_ * 2, 256, 0, stream>>>(parent, childW, childb, buf0);
    }
    // GIN layer 1
    gin_agg_kernel<<<F_ / 256, 256, 0, stream>>>(buf0, buf1);
    gemm_wmma_kernel<<<GN / 16, 128, 0, stream>>>(buf1, W1a, b1a, buf2, 1);
    gemm_wmma_kernel<<<GN / 16, 128, 0, stream>>>(buf2, W1b, b1b, buf0, 1); // outer relu
    bn_kernel<<<F_ / 256, 256, 0, stream>>>(buf0, g1, be1, buf1);
    // GIN layer 2
    gin_agg_kernel<<<F_ / 256, 256, 0, stream>>>(buf1, buf2);
    gemm_wmma_kernel<<<GN / 16, 128, 0, stream>>>(buf2, W2a, b2a, buf0, 1);
    gemm_wmma_kernel<<<GN / 16, 128, 0, stream>>>(buf0, W2b, b2b, buf1, 1); // outer relu
    bn_kernel<<<F_ / 256, 256, 0, stream>>>(buf1, g2, be2, out);
}